// BandSplit_91173565760184
// MI455X (gfx1250) — compile-verified
//
#include <hip/hip_runtime.h>

// BandSplit for MI455X (gfx1250, wave32, WMMA).
//
// Per band k: out[b,t,k,:] = gather(x)[b,t,k,(c,w)] . pre_w[k] + pre_b[k]
// => batched GEMM  M=B*T x K=C*Wmax x N=128 per band (64 bands), fp32.
// fp32 reference -> V_WMMA_F32_16X16X4_F32 keeps fp32 accumulation.
//
// Block = 256 threads (8 wave32), M-tile = 64 rows, N covered by 8 waves x 16.
// Gather stage: per-lane GLOBAL_LOAD_ASYNC_TO_LDS_B32 (ASYNCcnt path) writes
// the 64 x (C*Wmax) A-tile straight into LDS (row stride padded +4 dwords to
// kill the 16-way bank conflict when Wmax % 16 == 0), s_wait_asynccnt+barrier.
// MAC stage: each wave runs FOUR independent wmma chains (rows 0-15..48-63)
// sharing one B fragment per K-step -> 4x B reuse + deep ILP over L2 latency.

typedef __attribute__((ext_vector_type(2))) float v2f;
typedef __attribute__((ext_vector_type(8))) float v8f;

#define C_IN    4
#define D_OUT   128
#define T_LEN   1024
#define F_BINS  1025
#define K_BANDS 64
#define M_TILE  64
#define LDS_PAD 4          // dwords of row padding -> conflict-free ds loads

__global__ __launch_bounds__(256) void bandsplit_wmma_f32(
    const float* __restrict__ x,        // (B, C, T, F)
    const float* __restrict__ pre_w,    // (K, C*Wmax, D)
    const float* __restrict__ pre_b,    // (K, D)
    const int*   __restrict__ idx,      // (K, Wmax)
    float*       __restrict__ out,      // (B, D, T, K)
    int Wmax)
{
    extern __shared__ float smem[];               // M_TILE rows x S floats
    const int I    = C_IN * Wmax;                 // reduction dim
    const int S    = I + LDS_PAD;                 // padded LDS row stride
    const int k    = blockIdx.y;                  // band
    const int bt0  = blockIdx.x * M_TILE;         // first (b,t) row of tile
    const int b    = bt0 / T_LEN;                 // T_LEN % M_TILE == 0
    const int t0   = bt0 - b * T_LEN;
    const int tid  = threadIdx.x;

    // ---- async gather: copy x[b, c, t0+r, idx[k,w]] into LDS tile row r, column c*Wmax+w ----
    // Padded index entries are 0 (valid address); their pre_w columns are
    // zero, so the contribution is exactly 0 -> mask not needed.
    const int* idxk = idx + k * Wmax;
    for (int j = tid; j < M_TILE * I; j += 256) {
        const int r   = j / I;
        const int i   = j - r * I;
        const int c   = i / Wmax;
        const int w   = i - c * Wmax;
        const int col = idxk[w];
        const float* src =
            &x[(((size_t)b * C_IN + c) * T_LEN + (t0 + r)) * F_BINS + col];
        const unsigned lds_off = (unsigned)(r * S + i) * 4u; // dyn LDS base = 0
        asm volatile("global_load_async_to_lds_b32 %0, %1, off"
                     :: "v"(lds_off), "v"((unsigned long long)(size_t)src)
                     : "memory");
    }
    asm volatile("s_wait_asynccnt 0x0" ::: "memory");
    __syncthreads();

    // ---- per-wave quad 16x16 WMMA chains over d-cols [d0, d0+16) ----
    const int lane = tid & 31;
    const int wave = tid >> 5;          // 0..7 -> covers D=128
    const int n    = lane & 15;         // tile col / A row (M)
    const int koff = (lane >> 4) << 1;  // lanes 0-15: K=0,1 ; lanes 16-31: K=2,3
    const int d0   = wave << 4;
    const float* __restrict__ Wk = pre_w + (size_t)k * I * D_OUT;

    v8f acc[4] = {};                    // rows bt0+16*j .. bt0+16*j+15
    #pragma unroll 2
    for (int i0 = 0; i0 < I; i0 += 4) {
        // B fragment (4x16 f32), shared by all four wmmas
        const float* wp = Wk + (size_t)(i0 + koff) * D_OUT + d0 + n;
        v2f bf;
        bf.x = wp[0];
        bf.y = wp[D_OUT];
        // A fragments (16x4 f32) for the four row quarters
        #pragma unroll
        for (int j = 0; j < 4; ++j) {
            v2f a;
            a.x = smem[(n + 16 * j) * S + i0 + koff];
            a.y = smem[(n + 16 * j) * S + i0 + koff + 1];
            // 8 args: (neg_a, A, neg_b, B, c_mod, C, reuse_a, reuse_b)
            acc[j] = __builtin_amdgcn_wmma_f32_16x16x4_f32(
                         false, a, false, bf, (short)0, acc[j], false, false);
        }
    }

    // ---- bias + store into out[b, d, t, k] ----
    const float bias  = pre_b[k * D_OUT + d0 + n];
    const int   mbase = (lane >> 4) << 3;   // VGPR i: M=i (lanes 0-15) / M=i+8
    const size_t obase =
        (((size_t)b * D_OUT + (d0 + n)) * T_LEN + t0) * K_BANDS + k;
    #pragma unroll
    for (int j = 0; j < 4; ++j) {
        #pragma unroll
        for (int i = 0; i < 8; ++i) {
            out[obase + (size_t)(16 * j + mbase + i) * K_BANDS] =
                acc[j][i] + bias;
        }
    }
}

extern "C" void kernel_launch(void* const* d_in, const int* in_sizes, int n_in,
                              void* d_out, int out_size, void* d_ws, size_t ws_size,
                              hipStream_t stream) {
    const float* x     = (const float*)d_in[0];   // (B, C, T, F) f32
    const float* pre_w = (const float*)d_in[1];   // (K, C*Wmax, D) f32
    const float* pre_b = (const float*)d_in[2];   // (K, D) f32
    const int*   idx   = (const int*)d_in[3];     // (K, Wmax) i32
    // d_in[4] = mask, not needed (pre_w is zero-padded identically)

    const int Wmax = in_sizes[3] / K_BANDS;
    const int Bsz  = in_sizes[0] / (C_IN * T_LEN * F_BINS);
    const int I    = C_IN * Wmax;

    dim3 grid((Bsz * T_LEN) / M_TILE, K_BANDS);
    const size_t smem_bytes = (size_t)M_TILE * (I + LDS_PAD) * sizeof(float);

    bandsplit_wmma_f32<<<grid, 256, smem_bytes, stream>>>(
        x, pre_w, pre_b, idx, (float*)d_out, Wmax);
}